// ExpansiveLayer_20263655702652
// MI455X (gfx1250) — compile-verified
//
#include <hip/hip_runtime.h>

typedef float v2f __attribute__((ext_vector_type(2)));
typedef float v8f __attribute__((ext_vector_type(8)));

static constexpr int N_POOL   = 100000;
static constexpr int N_UNPOOL = 200000;
static constexpr int NE       = 1200000;
static constexpr int C_DIM    = 64;
static constexpr int ROWS     = 32;   // rows of output per block
static constexpr int AST      = 66;   // LDS stride for A tile (pad vs 64 banks, keeps 8B align)
static constexpr int WST      = 66;   // LDS stride for W^T tile

// ---------------------------------------------------------------------------
// GEMM1: h1_pre = ((x / score)[cluster]) @ W1 + b1 ; accumulate BN stats
// ---------------------------------------------------------------------------
__global__ __launch_bounds__(256) void gemm1_kernel(
    const float* __restrict__ x, const int* __restrict__ cluster,
    const float* __restrict__ score, const float* __restrict__ W,
    const float* __restrict__ bias, float* __restrict__ out,
    float* __restrict__ stat_sum, float* __restrict__ stat_sq)
{
    __shared__ float As[ROWS * AST];
    __shared__ float Wt[C_DIM * WST];
    __shared__ float lsum[C_DIM];
    __shared__ float lsq[C_DIM];

    const int tid = threadIdx.x;
    const int rowbase = blockIdx.x * ROWS;

    // Stage W transposed: Wt[col][k] = W[k][col]  (so B frags are contiguous pairs)
    for (int i = tid; i < C_DIM * C_DIM; i += 256) {
        int r = i >> 6, c = i & 63;
        Wt[c * WST + r] = W[i];
    }
    if (tid < C_DIM) { lsum[tid] = 0.f; lsq[tid] = 0.f; }

    // Gather 32 rows: As[rr][c] = x[cluster[row]][c] / score[cluster[row]]
    {
        int rr = tid >> 3;              // 0..31
        int c0 = (tid & 7) * 8;         // 0,8,..,56
        int gr = rowbase + rr;          // grid sized exactly: gr < N_UNPOOL
        int src = cluster[gr];
        float inv = 1.0f / score[src];
        const float* xp = x + (size_t)src * C_DIM + c0;
        #pragma unroll
        for (int k = 0; k < 8; ++k)
            As[rr * AST + c0 + k] = xp[k] * inv;
    }
    __syncthreads();

    const int wave = tid >> 5;          // 0..7
    const int lane = tid & 31;
    const int rt = wave >> 2;           // row tile 0..1
    const int ct = wave & 3;            // col tile 0..3
    const int half = lane >> 4;         // 0: K pair (0,1), 1: K pair (2,3)
    const int l16 = lane & 15;

    v8f acc = {};
    const float* Ap = &As[(rt * 16 + l16) * AST + 2 * half];
    const float* Bp = &Wt[(ct * 16 + l16) * WST + 2 * half];
    #pragma unroll
    for (int k = 0; k < 16; ++k) {
        v2f a = *(const v2f*)(Ap + 4 * k);
        v2f b = *(const v2f*)(Bp + 4 * k);
        acc = __builtin_amdgcn_wmma_f32_16x16x4_f32(false, a, false, b,
                                                    (short)0, acc, false, false);
    }

    // D layout: VGPR v -> row rt*16 + v + half*8, col ct*16 + l16
    const int col = ct * 16 + l16;
    const float bv = bias[col];
    float psum = 0.f, psq = 0.f;
    #pragma unroll
    for (int v = 0; v < 8; ++v) {
        float val = acc[v] + bv;
        size_t row = (size_t)(rowbase + rt * 16 + half * 8 + v);
        out[row * C_DIM + col] = val;
        psum += val;
        psq  += val * val;
    }
    atomicAdd(&lsum[col], psum);
    atomicAdd(&lsq[col],  psq);
    __syncthreads();
    if (tid < C_DIM) {
        atomicAdd(&stat_sum[tid], lsum[tid]);
        atomicAdd(&stat_sq[tid],  lsq[tid]);
    }
}

// ---------------------------------------------------------------------------
// GEMM2: hw = relu(a1*h1_pre + c1) @ Wg   (BN+ReLU fused into A staging)
// ---------------------------------------------------------------------------
__global__ __launch_bounds__(256) void gemm2_kernel(
    const float* __restrict__ h1, const float* __restrict__ coef_a,
    const float* __restrict__ coef_c, const float* __restrict__ W,
    float* __restrict__ out)
{
    __shared__ float As[ROWS * AST];
    __shared__ float Wt[C_DIM * WST];

    const int tid = threadIdx.x;
    const int rowbase = blockIdx.x * ROWS;

    for (int i = tid; i < C_DIM * C_DIM; i += 256) {
        int r = i >> 6, c = i & 63;
        Wt[c * WST + r] = W[i];
    }
    {
        int rr = tid >> 3;
        int c0 = (tid & 7) * 8;
        const float* hp = h1 + (size_t)(rowbase + rr) * C_DIM + c0;
        #pragma unroll
        for (int k = 0; k < 8; ++k) {
            float v = coef_a[c0 + k] * hp[k] + coef_c[c0 + k];
            As[rr * AST + c0 + k] = v > 0.f ? v : 0.f;
        }
    }
    __syncthreads();

    const int wave = tid >> 5;
    const int lane = tid & 31;
    const int rt = wave >> 2;
    const int ct = wave & 3;
    const int half = lane >> 4;
    const int l16 = lane & 15;

    v8f acc = {};
    const float* Ap = &As[(rt * 16 + l16) * AST + 2 * half];
    const float* Bp = &Wt[(ct * 16 + l16) * WST + 2 * half];
    #pragma unroll
    for (int k = 0; k < 16; ++k) {
        v2f a = *(const v2f*)(Ap + 4 * k);
        v2f b = *(const v2f*)(Bp + 4 * k);
        acc = __builtin_amdgcn_wmma_f32_16x16x4_f32(false, a, false, b,
                                                    (short)0, acc, false, false);
    }
    const int col = ct * 16 + l16;
    #pragma unroll
    for (int v = 0; v < 8; ++v) {
        size_t row = (size_t)(rowbase + rt * 16 + half * 8 + v);
        out[row * C_DIM + col] = acc[v];
    }
}

// ---------------------------------------------------------------------------
// BN coefficient folding: y = a*x + c  with a = g*rstd, c = be - mu*a
// ---------------------------------------------------------------------------
__global__ void bncoef_kernel(const float* __restrict__ sum, const float* __restrict__ sq,
                              const float* __restrict__ gamma, const float* __restrict__ beta,
                              float invN, float* __restrict__ a, float* __restrict__ c)
{
    int ch = threadIdx.x;
    float mu  = sum[ch] * invN;
    float var = sq[ch] * invN - mu * mu;
    float ga  = gamma[ch] * rsqrtf(var + 1e-5f);
    a[ch] = ga;
    c[ch] = beta[ch] - mu * ga;
}

__global__ void degree_kernel(const int* __restrict__ dstI, float* __restrict__ deg, int n)
{
    int e = blockIdx.x * blockDim.x + threadIdx.x;
    if (e < n) atomicAdd(&deg[dstI[e]], 1.0f);
}

__global__ void dis_kernel(const float* __restrict__ deg, float* __restrict__ dis, int n)
{
    int i = blockIdx.x * blockDim.x + threadIdx.x;
    if (i < n) dis[i] = rsqrtf(deg[i] + 1.0f);   // +1 = self loop
}

// out[i][c] = hw[i][c] * dis[i]^2   (self-loop contribution seeds accumulator)
__global__ void selfloop_kernel(const float* __restrict__ hw, const float* __restrict__ dis,
                                float* __restrict__ out, int total)
{
    int t = blockIdx.x * blockDim.x + threadIdx.x;
    if (t < total) {
        float d = dis[t >> 6];
        out[t] = hw[t] * d * d;
    }
}

__global__ void scatter_kernel(const float* __restrict__ hw, const float* __restrict__ dis,
                               const int* __restrict__ srcI, const int* __restrict__ dstI,
                               float* __restrict__ out, int total)
{
    int t = blockIdx.x * blockDim.x + threadIdx.x;
    if (t >= total) return;
    int e = t >> 6, c = t & 63;
    int s = srcI[e], d = dstI[e];
    atomicAdd(&out[(size_t)d * C_DIM + c],
              hw[(size_t)s * C_DIM + c] * dis[s] * dis[d]);
}

// BN2 stats over (out + bg); each thread owns a fixed channel (256 % 64 == 0)
__global__ __launch_bounds__(256) void stats2_kernel(const float* __restrict__ out,
                                                     const float* __restrict__ bg,
                                                     float* __restrict__ sum,
                                                     float* __restrict__ sq, int total)
{
    __shared__ float lsum[C_DIM];
    __shared__ float lsq[C_DIM];
    int tid = threadIdx.x;
    if (tid < C_DIM) { lsum[tid] = 0.f; lsq[tid] = 0.f; }
    __syncthreads();
    int c = tid & 63;
    float bv = bg[c];
    float ps = 0.f, pq = 0.f;
    int base = blockIdx.x * 4096;
    #pragma unroll
    for (int k = 0; k < 16; ++k) {
        int t = base + k * 256 + tid;
        if (t < total) {
            float v = out[t] + bv;
            ps += v;
            pq += v * v;
        }
    }
    atomicAdd(&lsum[c], ps);
    atomicAdd(&lsq[c],  pq);
    __syncthreads();
    if (tid < C_DIM) {
        atomicAdd(&sum[tid], lsum[tid]);
        atomicAdd(&sq[tid],  lsq[tid]);
    }
}

__global__ void final_kernel(float* __restrict__ out, const float* __restrict__ bg,
                             const float* __restrict__ a, const float* __restrict__ c,
                             int total)
{
    int t = blockIdx.x * blockDim.x + threadIdx.x;
    if (t < total) {
        int ch = t & 63;
        float v = a[ch] * (out[t] + bg[ch]) + c[ch];
        out[t] = v > 0.f ? v : 0.f;
    }
}

__global__ void copyint_kernel(const int* __restrict__ src, float* __restrict__ dst, int n)
{
    int t = blockIdx.x * blockDim.x + threadIdx.x;
    if (t < n) dst[t] = (float)src[t];
}

// ---------------------------------------------------------------------------
extern "C" void kernel_launch(void* const* d_in, const int* in_sizes, int n_in,
                              void* d_out, int out_size, void* d_ws, size_t ws_size,
                              hipStream_t stream)
{
    (void)in_sizes; (void)n_in; (void)out_size; (void)ws_size;

    const float* x       = (const float*)d_in[0];
    const int*   edge    = (const int*)  d_in[1];   // [2, E] row-major
    const int*   batch   = (const int*)  d_in[2];
    const int*   cluster = (const int*)  d_in[3];
    const float* score   = (const float*)d_in[4];
    const float* W1      = (const float*)d_in[5];
    const float* b1      = (const float*)d_in[6];
    const float* g1      = (const float*)d_in[7];
    const float* be1     = (const float*)d_in[8];
    const float* Wg      = (const float*)d_in[9];
    const float* bg      = (const float*)d_in[10];
    const float* g2      = (const float*)d_in[11];
    const float* be2     = (const float*)d_in[12];

    float* ws  = (float*)d_ws;
    float* h1  = ws;                       // 12,800,000
    float* hw  = ws + 12800000;            // 12,800,000
    float* deg = ws + 25600000;            // 200,000
    float* dis = ws + 25800000;            // 200,000
    float* st  = ws + 26000000;            // 512
    float *sum1 = st,       *sq1 = st + 64,  *a1 = st + 128, *c1 = st + 192;
    float *sum2 = st + 256, *sq2 = st + 320, *a2 = st + 384, *c2 = st + 448;

    float* out    = (float*)d_out;         // [200000*64] result tensor
    float* out_ei = out + 12800000;        // edge_index pass-through (2*E)
    float* out_b  = out + 15200000;        // batch pass-through (N_UNPOOL)

    const int* srcI = edge;                // edge_index[0]
    const int* dstI = edge + NE;           // edge_index[1]

    const int TOTAL = N_UNPOOL * C_DIM;    // 12,800,000
    const float invN = 1.0f / (float)N_UNPOOL;

    hipMemsetAsync(deg, 0, N_UNPOOL * sizeof(float), stream);
    hipMemsetAsync(st,  0, 512 * sizeof(float), stream);

    gemm1_kernel<<<N_UNPOOL / ROWS, 256, 0, stream>>>(x, cluster, score, W1, b1,
                                                      h1, sum1, sq1);
    bncoef_kernel<<<1, 64, 0, stream>>>(sum1, sq1, g1, be1, invN, a1, c1);
    gemm2_kernel<<<N_UNPOOL / ROWS, 256, 0, stream>>>(h1, a1, c1, Wg, hw);

    degree_kernel<<<(NE + 255) / 256, 256, 0, stream>>>(dstI, deg, NE);
    dis_kernel<<<(N_UNPOOL + 255) / 256, 256, 0, stream>>>(deg, dis, N_UNPOOL);
    selfloop_kernel<<<TOTAL / 256, 256, 0, stream>>>(hw, dis, out, TOTAL);
    scatter_kernel<<<(NE * C_DIM) / 256, 256, 0, stream>>>(hw, dis, srcI, dstI, out,
                                                           NE * C_DIM);
    stats2_kernel<<<TOTAL / 4096, 256, 0, stream>>>(out, bg, sum2, sq2, TOTAL);
    bncoef_kernel<<<1, 64, 0, stream>>>(sum2, sq2, g2, be2, invN, a2, c2);
    final_kernel<<<TOTAL / 256, 256, 0, stream>>>(out, bg, a2, c2, TOTAL);

    copyint_kernel<<<(2 * NE + 255) / 256, 256, 0, stream>>>(edge, out_ei, 2 * NE);
    copyint_kernel<<<(N_UNPOOL + 255) / 256, 256, 0, stream>>>(batch, out_b, N_UNPOOL);
}